// MyBatchNormList_3813930959608
// MI455X (gfx1250) — compile-verified
//
#include <hip/hip_runtime.h>
#include <math.h>

// ---------------------------------------------------------------------------
// MyBatchNormList for MI455X (gfx1250, wave32).
//
// Roofline: 151 MB in + 151 MB out, trivial FLOPs -> HBM bound (~13 us floor
// @ 23.3 TB/s). Inputs (151 MB) fit in the 192 MB L2, so the two-pass scheme
// costs ~one HBM read + one HBM write (the pass-2 re-read hits L2).
//
// 3 launches total:
//   1) bn_partial_kernel: per-channel partial moments -> ws (register accum,
//      no atomics, M compile-time specialized per level).
//   2) bn_stats_kernel:   reduce 16 partials/channel via chained
//      V_WMMA_F32_16X16X4_F32 (row-sum as matmul-with-ones), then
//      inv = 1/max(EPS, 0.5*std + 0.5*batch_std).
//   3) bn_scale_all_kernel: single fused float4 streaming multiply over all
//      six levels (b128 loads/stores, compile-time M -> magic-number divides).
// ---------------------------------------------------------------------------

typedef __attribute__((ext_vector_type(2))) float v2f;
typedef __attribute__((ext_vector_type(8))) float v8f;

#define NLVL  6
#define NCH   512
#define BATCH 1024
#define PB    16      // partial blocks per l
#define BPB   64      // batches per partial block (PB*BPB == BATCH)
#define EPSV  1e-5f
#define QPL   262144u // float4 quads per unit M: BATCH*NCH*2*M floats / 4 / M

struct Ptr6 { const float* p[NLVL]; };

// ---------------------------------------------------------------------------
// Pass 1a: one thread per channel, 64 batches per block slice.
// Thread c reads 2*M contiguous floats per batch; the unrolled m-loop forms
// one load clause covering a contiguous 256*M-byte region per wave per batch,
// so cachelines are fully consumed. Register accumulation -> no extra traffic.
// ---------------------------------------------------------------------------
template <int M>
__device__ __forceinline__ void partial_body(const float2* __restrict__ f2,
                                             int c, int b0, float& s, float& q) {
    for (int i = 0; i < BPB; ++i) {
        const float2* __restrict__ row = f2 + (size_t)((b0 + i) * NCH + c) * M;
        #pragma unroll
        for (int m = 0; m < M; ++m) {
            float2 v = row[m];
            float n2 = v.x * v.x + v.y * v.y;   // |z|^2: second moment directly
            q += n2;
            s += sqrtf(n2);
        }
    }
}

__global__ __launch_bounds__(NCH)
void bn_partial_kernel(Ptr6 fs, float* __restrict__ partials) {
    const int l  = blockIdx.y;
    const int c  = threadIdx.x;                // 0..511
    const int b0 = blockIdx.x * BPB;
    const float2* __restrict__ f2 = (const float2*)fs.p[l];

    float s = 0.0f, q = 0.0f;
    switch (l) {                               // uniform scalar branch per block
        case 0: partial_body<1 >(f2, c, b0, s, q); break;
        case 1: partial_body<3 >(f2, c, b0, s, q); break;
        case 2: partial_body<5 >(f2, c, b0, s, q); break;
        case 3: partial_body<7 >(f2, c, b0, s, q); break;
        case 4: partial_body<9 >(f2, c, b0, s, q); break;
        default: partial_body<11>(f2, c, b0, s, q); break;
    }
    float* dst = partials + (size_t)(l * PB + blockIdx.x) * (2 * NCH);
    dst[c]       = s;
    dst[NCH + c] = q;
}

// ---------------------------------------------------------------------------
// Pass 1b: one wave per l. Reduce PB=16 partials for 16 channels at a time
// with 4 chained V_WMMA_F32_16X16X4_F32 (B = ones). Per the ISA layouts:
//   A (16x4 f32): lanes 0-15 hold K=0,1 in VGPR0/1; lanes 16-31 hold K=2,3.
//   D (16x16 f32): lane n (0-15) VGPR r = D[r][n]; lanes 16-31 = D[8+r][n].
// With B = ones every D column equals the per-channel total, so lanes 0 and 16
// each own 8 channel totals and finish the statistics.
// ---------------------------------------------------------------------------
__global__ __launch_bounds__(32)
void bn_stats_kernel(Ptr6 stds, const float* __restrict__ partials,
                     float* __restrict__ invscale) {
    const int l = blockIdx.x;
    const int M = 2 * l + 1;
    const float invN = 1.0f / (float)(BATCH * M);
    const float* __restrict__ base = partials + (size_t)l * PB * (2 * NCH);
    const float* __restrict__ stdv = stds.p[l];
    const int lane = threadIdx.x;
    const int m  = lane & 15;              // A-matrix row (channel offset)
    const int kb = (lane >> 4) << 1;       // K base: 0 (lanes 0-15) / 2 (16-31)

    v2f ones; ones.x = 1.0f; ones.y = 1.0f;

    for (int c0 = 0; c0 < NCH; c0 += 16) {
        v8f accS = {};
        v8f accQ = {};
        #pragma unroll
        for (int g = 0; g < 4; ++g) {      // 4 WMMAs cover all 16 partials
            const int p0 = g * 4 + kb;
            const float* r0 = base + (size_t)p0 * (2 * NCH);
            const float* r1 = r0 + 2 * NCH;
            v2f aS, aQ;
            aS.x = r0[c0 + m];        aS.y = r1[c0 + m];
            aQ.x = r0[NCH + c0 + m];  aQ.y = r1[NCH + c0 + m];
            accS = __builtin_amdgcn_wmma_f32_16x16x4_f32(
                false, aS, false, ones, (short)0, accS, false, false);
            accQ = __builtin_amdgcn_wmma_f32_16x16x4_f32(
                false, aQ, false, ones, (short)0, accQ, false, false);
        }
        if ((lane & 15) == 0) {            // lanes 0 and 16 hold column N=0
            const int ch = c0 + ((lane >> 4) << 3);
            #pragma unroll
            for (int r = 0; r < 8; ++r) {
                float s    = accS[r];
                float q    = accQ[r];
                float mean = s * invN;
                float var  = q * invN - mean * mean;
                float bstd = sqrtf(fmaxf(var, 0.0f));
                float ns   = 0.5f * stdv[ch + r] + 0.5f * bstd;  // CNT = 1.0
                float sc   = fmaxf(EPSV, ns);
                invscale[l * NCH + ch + r] = 1.0f / sc;
            }
        }
    }
}

// ---------------------------------------------------------------------------
// Pass 2 (fused, single launch): float4 (b128) streaming multiply.
// Output quad base for level l is QPL*l^2 (since sum of (2k+1) = l^2).
// A float4 = 2 complex pairs; channel of pair p is (p/M) & 511 with M a
// compile-time constant per unrolled level -> magic-number mulhi, no div.
// 1/scale for all levels cached in 12 KB of LDS.
// ---------------------------------------------------------------------------
template <int L>
__device__ __forceinline__ void scale_level(const float4* __restrict__ fin,
                                            float4* __restrict__ out,
                                            const float* __restrict__ sInv,
                                            unsigned gid, unsigned stride) {
    constexpr unsigned M  = 2u * L + 1u;
    const unsigned nq     = QPL * M;               // quads in this level
    const unsigned qBase  = QPL * (unsigned)(L * L);
    const float* __restrict__ inv = sInv + L * NCH;

    for (unsigned q = gid; q < nq; q += stride) {
        float4 v = fin[q];
        const unsigned p0 = 2u * q;
        const unsigned c0 = (p0 / M) & (NCH - 1);        // compile-time M
        const unsigned c1 = ((p0 + 1u) / M) & (NCH - 1);
        const float s0 = inv[c0];
        const float s1 = inv[c1];
        v.x *= s0; v.y *= s0;
        v.z *= s1; v.w *= s1;
        out[qBase + q] = v;
    }
}

__global__ __launch_bounds__(256)
void bn_scale_all_kernel(Ptr6 fs, const float* __restrict__ invscale,
                         float4* __restrict__ out) {
    __shared__ float sInv[NLVL * NCH];
    for (int i = threadIdx.x; i < NLVL * NCH; i += 256) sInv[i] = invscale[i];
    __syncthreads();

    const unsigned gid    = blockIdx.x * blockDim.x + threadIdx.x;
    const unsigned stride = gridDim.x * blockDim.x;

    scale_level<0>((const float4*)fs.p[0], out, sInv, gid, stride);
    scale_level<1>((const float4*)fs.p[1], out, sInv, gid, stride);
    scale_level<2>((const float4*)fs.p[2], out, sInv, gid, stride);
    scale_level<3>((const float4*)fs.p[3], out, sInv, gid, stride);
    scale_level<4>((const float4*)fs.p[4], out, sInv, gid, stride);
    scale_level<5>((const float4*)fs.p[5], out, sInv, gid, stride);
}

// ---------------------------------------------------------------------------
// Inputs: d_in[0..5] = f0..f5 (fp32), d_in[6..11] = stds0..stds5 (fp32).
// Output: f0/s0 .. f5/s5 concatenated flat.
// Workspace: [ partials: 6*16*1024 floats | invscale: 6*512 floats ] ~396 KB.
// Everything in ws is written before it is read -> no init needed, and the
// pipeline is fully deterministic.
// ---------------------------------------------------------------------------
extern "C" void kernel_launch(void* const* d_in, const int* in_sizes, int n_in,
                              void* d_out, int out_size, void* d_ws, size_t ws_size,
                              hipStream_t stream) {
    (void)in_sizes; (void)n_in; (void)out_size; (void)ws_size;

    Ptr6 fs, st;
    for (int l = 0; l < NLVL; ++l) {
        fs.p[l] = (const float*)d_in[l];
        st.p[l] = (const float*)d_in[NLVL + l];
    }

    float* wsf      = (float*)d_ws;
    float* partials = wsf;                             // 6*16*2*512 floats
    float* invscale = wsf + (size_t)NLVL * PB * 2 * NCH;

    bn_partial_kernel<<<dim3(PB, NLVL), NCH, 0, stream>>>(fs, partials);
    bn_stats_kernel<<<NLVL, 32, 0, stream>>>(st, partials, invscale);
    bn_scale_all_kernel<<<4096, 256, 0, stream>>>(fs, invscale, (float4*)d_out);
}